// ComplexEMA_8890582302873
// MI455X (gfx1250) — compile-verified
//
#include <hip/hip_runtime.h>
#include <math.h>

// Problem constants (from reference): B=2, D=1024, N=16, L=4096
#define B_  2
#define D_  1024
#define N_  16
#define L_  4096
#define LK_PAD 16      // zero pad in front of k  (handles r-r' < 0 at m=0)
#define LX_PAD 1024    // 64 zero tile-columns in front of x (handles i-m < 0)

typedef float v2f __attribute__((ext_vector_type(2)));
typedef float v8f __attribute__((ext_vector_type(8)));
typedef int   v4i __attribute__((ext_vector_type(4)));

// Pointer types for the async global->LDS DMA builtin (prototype learned from
// the compiler diagnostic: arg0 is int4* in addrspace(1)).
typedef __attribute__((address_space(1))) v4i* glb_v4i_p;
typedef __attribute__((address_space(3))) v4i* lds_v4i_p;

// CDNA5 async global->LDS copy path (ASYNCcnt), guarded so absence of the
// builtin falls back to plain loads without breaking the build.
#if defined(__has_builtin)
#if __has_builtin(__builtin_amdgcn_global_load_async_to_lds_b128) && \
    __has_builtin(__builtin_amdgcn_s_wait_asynccnt)
#define USE_ASYNC_LDS 1
#endif
#endif
#ifndef USE_ASYNC_LDS
#define USE_ASYNC_LDS 0
#endif

// --------------------------------------------------------------------------
// Kernel 1: build k[d,l] = Re( sum_n gp[d,n] * q[d,n]^l ),  q = radius*e^{i phi}
// One block per d, 256 threads, each thread produces 16 l-values.
// --------------------------------------------------------------------------
__global__ __launch_bounds__(256)
void ema_build_kernel(const float* __restrict__ alpha,
                      const float* __restrict__ delta,
                      const float* __restrict__ theta,
                      const float* __restrict__ gr,
                      const float* __restrict__ gi,
                      float* __restrict__ kbuf)
{
    const int d   = blockIdx.x;
    const int tid = threadIdx.x;
    const float scale = 0.25f;                       // 1/sqrt(N), N=16

    float th = theta[d];
    th = 1.0f / (1.0f + __expf(-th));
    th *= (2.0f * 3.14159265358979323846f / (float)N_);

    float gpr[N_], gpi[N_], lr[N_], phi[N_];
#pragma unroll
    for (int n = 0; n < N_; ++n) {
        float a  = alpha[d * N_ + n];
        float de = delta[d * N_ + n];
        a  = 1.0f / (1.0f + __expf(-a));
        de = 1.0f / (1.0f + __expf(-de));
        float radius = fminf(1.0f - a * de, 1.0f);   // in (0,1)
        lr[n]  = __logf(radius);
        phi[n] = (float)(n + 1) * th;
        gpr[n] = gr[d * N_ + n] * scale * a;
        gpi[n] = gi[d * N_ + n] * scale * a;
    }

#pragma unroll
    for (int j = 0; j < L_ / 256; ++j) {
        const int   l  = tid + 256 * j;
        const float fl = (float)l;
        float acc = 0.0f;
#pragma unroll
        for (int n = 0; n < N_; ++n) {
            float m = __expf(fl * lr[n]);            // radius^l
            float s, c;
            __sincosf(fl * phi[n], &s, &c);
            acc += gpr[n] * (m * c) - gpi[n] * (m * s);
        }
        kbuf[d * L_ + l] = acc;
    }
}

// --------------------------------------------------------------------------
// Kernel 2: causal convolution y[b,d,:] = k[d,:] (*) x[b,d,:]  + omega[d]*x
// Block-Toeplitz with V_WMMA_F32_16X16X4_F32 (full fp32 precision).
//
//   Y[r, i] = sum_{m<=i} sum_{r'} A^{(m)}[r,r'] * X[r', i-m]
//   A^{(m)}[r,r'] = k[16m + r - r']         (zero-pad handles causality)
//
// One WG (128 threads, 4 waves) per (b,d). Wave w owns 4 C tiles covering
// tile-columns [64w, 64w+64). Per m-step the A tile is loaded to registers
// ONCE and reused for all 4 B tiles -> 16 wmma per (1 A + 4 B) KB of LDS
// traffic (0.31 KB/wmma). The m-loop is peeled into 4 phases whose active
// tile set is static, and the wave id is read-first-laned so all loop
// control is scalar (s_cmp/s_cbranch) and EXEC is never masked around WMMA.
// --------------------------------------------------------------------------
__global__ __launch_bounds__(128)
void ema_conv_kernel(const float* __restrict__ x,
                     const float* __restrict__ kbuf,
                     const float* __restrict__ omega,
                     float* __restrict__ out)
{
    __shared__ float lk[LK_PAD + L_];   // 16448 B
    __shared__ float lx[LX_PAD + L_];   // 20480 B

    const int d   = blockIdx.x;
    const int b   = blockIdx.y;
    const int tid = threadIdx.x;

    const float* xp = x    + ((size_t)b * D_ + d) * L_;
    const float* kp = kbuf + (size_t)d * L_;

    // ---- zero the pads ----
    if (tid < LK_PAD) lk[tid] = 0.0f;
#pragma unroll
    for (int j = 0; j < LX_PAD / 128; ++j) lx[tid + 128 * j] = 0.0f;

    // ---- stage k and x into LDS ----
#if USE_ASYNC_LDS
    // 16 B per lane per issue, 8 issues per array: ASYNCcnt-tracked DMA that
    // bypasses VGPRs entirely (global_load_async_to_lds_b128).
#pragma unroll
    for (int j = 0; j < L_ / (128 * 4); ++j) {
        const int e = tid * 4 + 512 * j;
        __builtin_amdgcn_global_load_async_to_lds_b128(
            (glb_v4i_p)(kp + e), (lds_v4i_p)&lk[LK_PAD + e], 0, 0);
        __builtin_amdgcn_global_load_async_to_lds_b128(
            (glb_v4i_p)(xp + e), (lds_v4i_p)&lx[LX_PAD + e], 0, 0);
    }
    __builtin_amdgcn_s_wait_asynccnt(0);
#else
#pragma unroll
    for (int j = 0; j < L_ / 128; ++j) {
        lk[LK_PAD + tid + j * 128] = kp[tid + j * 128];
        lx[LX_PAD + tid + j * 128] = xp[tid + j * 128];
    }
#endif
    __syncthreads();

    const int lane = tid & 31;
    const int n16  = lane & 15;       // A: row M=r ; B: column N=n
    const int half = lane >> 4;       // 0 or 1
    const int c0   = half * 2;        // K sub-lane offset per ISA A/B layout
    // Wave-uniform tile base, forced into an SGPR -> scalar loop control.
    const int t0   = __builtin_amdgcn_readfirstlane(tid >> 5) * 64;

    v8f acc0 = {}, acc1 = {}, acc2 = {}, acc3 = {};

    // One m-step: load A^{(m)} once, apply to tiles c = CMIN..3 (branch-free).
#define EMA_STEP(CMIN)                                                        \
    {                                                                         \
        const int kbase = 16 * m + n16 + LK_PAD;                              \
        v2f a[4];                                                             \
        _Pragma("unroll")                                                     \
        for (int kk = 0; kk < 4; ++kk) {                                      \
            const int aidx = kbase - (4 * kk + c0);                           \
            a[kk].x = lk[aidx];        /* k[16m + r - (4kk+c0)]   */          \
            a[kk].y = lk[aidx - 1];    /* k[16m + r - (4kk+c0+1)] */          \
        }                                                                     \
        _Pragma("unroll")                                                     \
        for (int c = (CMIN); c < 4; ++c) {                                    \
            const int colbase = t0 + 16 * c + n16 - m;  /* may hit zero pad */\
            v8f av = (c == 0) ? acc0 : (c == 1) ? acc1                        \
                   : (c == 2) ? acc2 : acc3;                                  \
            _Pragma("unroll")                                                 \
            for (int kk = 0; kk < 4; ++kk) {                                  \
                const v2f bpk =                                               \
                    *(const v2f*)&lx[LX_PAD + colbase * 16 + 4 * kk + c0];    \
                av = __builtin_amdgcn_wmma_f32_16x16x4_f32(                   \
                         false, a[kk], false, bpk, (short)0, av,              \
                         false, false);                                       \
            }                                                                 \
            if (c == 0) acc0 = av; else if (c == 1) acc1 = av;                \
            else if (c == 2) acc2 = av; else acc3 = av;                       \
        }                                                                     \
    }

    // Triangular schedule: tile c is live while m <= t0 + 16c + 15.
    int m = 0;
    for (; m < t0 + 16; ++m) EMA_STEP(0)   // all 4 tiles
    for (; m < t0 + 32; ++m) EMA_STEP(1)   // tiles 1..3
    for (; m < t0 + 48; ++m) EMA_STEP(2)   // tiles 2..3
    for (; m < t0 + 64; ++m) EMA_STEP(3)   // tile 3
#undef EMA_STEP

    // ---- write out with residual omega[d]*x (x already in LDS) ----
    const float w = omega[d];
    float* yp = out + ((size_t)b * D_ + d) * L_;
#pragma unroll
    for (int c = 0; c < 4; ++c) {
        const v8f av = (c == 0) ? acc0 : (c == 1) ? acc1
                     : (c == 2) ? acc2 : acc3;
#pragma unroll
        for (int v = 0; v < 8; ++v) {
            const int r = v + 8 * half;                 // C/D: M = v + 8*half
            const int l = (t0 + 16 * c + n16) * 16 + r; // N = n16 tile-column
            yp[l] = av[v] + lx[LX_PAD + l] * w;
        }
    }
}

// --------------------------------------------------------------------------
extern "C" void kernel_launch(void* const* d_in, const int* in_sizes, int n_in,
                              void* d_out, int out_size, void* d_ws, size_t ws_size,
                              hipStream_t stream)
{
    const float* x     = (const float*)d_in[0];   // (B,D,L)
    const float* alpha = (const float*)d_in[1];   // (D,N,1)
    const float* delta = (const float*)d_in[2];   // (D,N,1)
    const float* theta = (const float*)d_in[3];   // (D,1,1)
    const float* gr    = (const float*)d_in[4];   // (D,N)
    const float* gi    = (const float*)d_in[5];   // (D,N)
    const float* omega = (const float*)d_in[6];   // (D,)
    float* kbuf = (float*)d_ws;                   // D*L fp32 = 16.8 MB scratch
    float* out  = (float*)d_out;                  // (B,D,L)

    ema_build_kernel<<<dim3(D_), dim3(256), 0, stream>>>(alpha, delta, theta,
                                                         gr, gi, kbuf);
    ema_conv_kernel<<<dim3(D_, B_), dim3(128), 0, stream>>>(x, kbuf, omega, out);
}